// PointNetSetAbstraction_5214090297616
// MI455X (gfx1250) — compile-verified
//
#include <hip/hip_runtime.h>

typedef float v2f __attribute__((ext_vector_type(2)));
typedef float v8f __attribute__((ext_vector_type(8)));

// Problem constants (from reference)
constexpr int B_ = 16, N_ = 4096, D_ = 64, S_ = 1024, K_ = 32;
constexpr long P_ = (long)B_ * S_ * K_;          // 524288 grouped points
constexpr int C0IN = 67, C0P = 68;               // layer0 in (padded to 68)
constexpr float BN_EPS = 1e-5f;

// ---------------- workspace layout (floats) ----------------
constexpr size_t OFF_X0    = 0;                          // P*68
constexpr size_t OFF_Y0    = OFF_X0 + (size_t)P_ * 68;   // P*64
constexpr size_t OFF_Y1    = OFF_X0;                     // alias dead X0
constexpr size_t OFF_Y2    = OFF_Y0 + (size_t)P_ * 64;   // P*128
constexpr size_t OFF_ST0   = OFF_Y2 + (size_t)P_ * 128;  // 128 (sum64+ssq64)
constexpr size_t OFF_ST1   = OFF_ST0 + 128;
constexpr size_t OFF_ST2   = OFF_ST1 + 128;              // 256
constexpr size_t OFF_NRM0  = OFF_ST2 + 256;              // 128 (scale64+shift64)
constexpr size_t OFF_NRM1  = OFF_NRM0 + 128;
constexpr size_t OFF_NRM2  = OFF_NRM1 + 128;             // 256

__global__ void zero_kernel(float* p, int n) {
    int i = blockIdx.x * blockDim.x + threadIdx.x;
    if (i < n) p[i] = 0.f;
}

// ---------------- FPS: one block per batch ----------------
__global__ __launch_bounds__(1024)
void fps_kernel(const float* __restrict__ xyz, float* __restrict__ outXYZ) {
    __shared__ float xs[N_], ys[N_], zs[N_], dist[N_];
    __shared__ float rv[32];
    __shared__ int   ri[32];
    __shared__ int   sfar;
    const int b = blockIdx.x, tid = threadIdx.x;
    const float* base = xyz + (size_t)b * N_ * 3;
    for (int i = tid; i < N_; i += 1024) {
        xs[i] = base[i * 3 + 0]; ys[i] = base[i * 3 + 1]; zs[i] = base[i * 3 + 2];
        dist[i] = 1e10f;
    }
    if (tid == 0) sfar = 0;
    __syncthreads();
    const int wid = tid >> 5, lane = tid & 31;
    for (int it = 0; it < S_; ++it) {
        const int far = sfar;
        const float cx = xs[far], cy = ys[far], cz = zs[far];
        if (tid == 0) {                       // emit previous 'far' (matches scan output)
            float* o = outXYZ + ((size_t)b * S_ + it) * 3;
            o[0] = cx; o[1] = cy; o[2] = cz;
        }
        float mv = -1.f; int mi = 0x7fffffff;
        #pragma unroll
        for (int j = 0; j < 4; ++j) {
            int i = j * 1024 + tid;
            float dx = xs[i] - cx, dy = ys[i] - cy, dz = zs[i] - cz;
            float d  = fmaf(dx, dx, fmaf(dy, dy, dz * dz));
            float nd = fminf(dist[i], d);
            dist[i] = nd;
            if (nd > mv) { mv = nd; mi = i; }
        }
        #pragma unroll
        for (int off = 16; off > 0; off >>= 1) {
            float ov = __shfl_xor(mv, off, 32); int oi = __shfl_xor(mi, off, 32);
            if (ov > mv || (ov == mv && oi < mi)) { mv = ov; mi = oi; }
        }
        if (lane == 0) { rv[wid] = mv; ri[wid] = mi; }
        __syncthreads();
        if (wid == 0) {
            mv = rv[lane]; mi = ri[lane];
            #pragma unroll
            for (int off = 16; off > 0; off >>= 1) {
                float ov = __shfl_xor(mv, off, 32); int oi = __shfl_xor(mi, off, 32);
                if (ov > mv || (ov == mv && oi < mi)) { mv = ov; mi = oi; }
            }
            if (lane == 0) sfar = mi;
        }
        __syncthreads();
    }
}

// ---------------- KNN + gather + concat -> X0 [P,68] ----------------
// grid = B*64 blocks; each block: 1 batch's cloud in LDS, handles 16 centroids.
__global__ __launch_bounds__(256)
void knn_group_kernel(const float* __restrict__ xyz, const float* __restrict__ pf,
                      const float* __restrict__ sampled, float* __restrict__ x0) {
    __shared__ float xs[N_], ys[N_], zs[N_];
    __shared__ float rv[8];
    __shared__ int   ri[8];
    __shared__ int   sel[K_];
    __shared__ int   sbc;
    const int b  = blockIdx.x >> 6;
    const int cg = blockIdx.x & 63;
    const int tid = threadIdx.x;
    const float* base = xyz + (size_t)b * N_ * 3;
    for (int i = tid; i < N_; i += 256) {
        xs[i] = base[i * 3]; ys[i] = base[i * 3 + 1]; zs[i] = base[i * 3 + 2];
    }
    __syncthreads();
    const int wid = tid >> 5, lane = tid & 31;
    for (int c = 0; c < 16; ++c) {
        const int bs = b * S_ + cg * 16 + c;
        const float qx = sampled[(size_t)bs * 3 + 0];
        const float qy = sampled[(size_t)bs * 3 + 1];
        const float qz = sampled[(size_t)bs * 3 + 2];
        float dl[16];
        #pragma unroll
        for (int j = 0; j < 16; ++j) {
            int i = j * 256 + tid;
            float dx = xs[i] - qx, dy = ys[i] - qy, dz = zs[i] - qz;
            dl[j] = fmaf(dx, dx, fmaf(dy, dy, dz * dz));
        }
        for (int r = 0; r < K_; ++r) {
            float mv = 3.4e38f; int mi = 0x7fffffff;
            #pragma unroll
            for (int j = 0; j < 16; ++j) {
                int i = j * 256 + tid;
                if (dl[j] < mv) { mv = dl[j]; mi = i; }
            }
            #pragma unroll
            for (int off = 16; off > 0; off >>= 1) {
                float ov = __shfl_xor(mv, off, 32); int oi = __shfl_xor(mi, off, 32);
                if (ov < mv || (ov == mv && oi < mi)) { mv = ov; mi = oi; }
            }
            if (lane == 0) { rv[wid] = mv; ri[wid] = mi; }
            __syncthreads();
            if (tid == 0) {
                float bv = rv[0]; int bi = ri[0];
                for (int w = 1; w < 8; ++w)
                    if (rv[w] < bv || (rv[w] == bv && ri[w] < bi)) { bv = rv[w]; bi = ri[w]; }
                sel[r] = bi; sbc = bi;
            }
            __syncthreads();
            const int win = sbc;
            if ((win & 255) == tid) {
                const int jj = win >> 8;
                #pragma unroll
                for (int j = 0; j < 16; ++j) if (j == jj) dl[j] = 3.4e38f;
            }
        }
        // gather: 32 rows x (3 rel-xyz + 64 feat + 1 zero-pad)
        for (int k = wid; k < K_; k += 8) {
            const int n = sel[k];
            float* dst = x0 + ((size_t)bs * K_ + k) * C0P;
            if (lane == 0) {
                dst[0] = xs[n] - qx; dst[1] = ys[n] - qy; dst[2] = zs[n] - qz;
                dst[67] = 0.f;
            }
            const float* f = pf + ((size_t)b * N_ + n) * D_ + 2 * lane;
            dst[3 + 2 * lane] = f[0];
            dst[4 + 2 * lane] = f[1];
        }
        __syncthreads();
    }
}

// ---------------- WMMA GEMM + (optional) fused BN/ReLU on input + stats ----------------
// X [P, CINP] row-major; W [COUT, CIN]; Y [P, COUT]; stats = [sum COUT][ssq COUT]
template <int CIN, int CINP, int COUT, bool BNIN>
__global__ __launch_bounds__(256)
void gemm_bn_kernel(const float* __restrict__ X, const float* __restrict__ W,
                    const float* __restrict__ bias, const float* __restrict__ normIn,
                    float* __restrict__ Y, float* __restrict__ stats) {
    constexpr int TN = COUT / 16;
    __shared__ float Wl[COUT * CINP];
    __shared__ float sIn[CINP], tIn[CINP];
    __shared__ float lsum[COUT], lsq[COUT];
    const int tid = threadIdx.x;
    for (int i = tid; i < COUT * CINP; i += 256) {
        int o = i / CINP, c = i % CINP;
        Wl[i] = (c < CIN) ? W[o * CIN + c] : 0.f;
    }
    if (BNIN) {
        for (int i = tid; i < CINP; i += 256) { sIn[i] = normIn[i]; tIn[i] = normIn[CINP + i]; }
    }
    for (int i = tid; i < COUT; i += 256) { lsum[i] = 0.f; lsq[i] = 0.f; }
    __syncthreads();

    const int wid = tid >> 5, lane = tid & 31;
    const int lr = lane & 15;
    const int koff = (lane >> 4) ? 2 : 0;            // A/B K-halves per ISA layout
    const long p0 = (long)blockIdx.x * 128 + wid * 16;

    v8f acc[TN];
    #pragma unroll
    for (int t = 0; t < TN; ++t) {
        const float bv = bias[t * 16 + lr];
        #pragma unroll
        for (int i = 0; i < 8; ++i) acc[t][i] = bv;
    }

    const float* xrow = X + (p0 + lr) * CINP;
    for (int kc = 0; kc < CINP; kc += 4) {
        v2f a = *(const v2f*)(xrow + kc + koff);
        if (BNIN) {
            a.x = fmaxf(fmaf(a.x, sIn[kc + koff],     tIn[kc + koff]),     0.f);
            a.y = fmaxf(fmaf(a.y, sIn[kc + koff + 1], tIn[kc + koff + 1]), 0.f);
        }
        #pragma unroll
        for (int t = 0; t < TN; ++t) {
            v2f bm = *(const v2f*)(&Wl[(t * 16 + lr) * CINP + kc + koff]);
            acc[t] = __builtin_amdgcn_wmma_f32_16x16x4_f32(
                false, a, false, bm, (short)0, acc[t], false, false);
        }
    }

    #pragma unroll
    for (int t = 0; t < TN; ++t) {
        const int ch = t * 16 + lr;
        float s = 0.f, s2 = 0.f;
        #pragma unroll
        for (int i = 0; i < 8; ++i) {
            const float v = acc[t][i];
            const long row = p0 + i + ((lane >= 16) ? 8 : 0);
            Y[row * COUT + ch] = v;
            s += v; s2 += v * v;
        }
        s  += __shfl_xor(s, 16, 32);
        s2 += __shfl_xor(s2, 16, 32);
        if (lane < 16) { atomicAdd(&lsum[ch], s); atomicAdd(&lsq[ch], s2); }
    }
    __syncthreads();
    for (int i = tid; i < COUT; i += 256) {
        atomicAdd(&stats[i], lsum[i]);
        atomicAdd(&stats[COUT + i], lsq[i]);
    }
}

// ---------------- stats -> (scale, shift) ----------------
__global__ void finalize_stats_kernel(const float* __restrict__ stats, const float* __restrict__ g,
                                      const float* __restrict__ beta, float* __restrict__ norm, int C) {
    const int i = threadIdx.x;
    if (i < C) {
        const float inv_n = 1.f / (float)P_;
        const float mu  = stats[i] * inv_n;
        const float var = stats[C + i] * inv_n - mu * mu;
        const float sc  = g[i] * rsqrtf(var + BN_EPS);
        norm[i] = sc;
        norm[C + i] = beta[i] - mu * sc;
    }
}

// ---------------- BN + ReLU + max over K -> d_out feats ----------------
__global__ __launch_bounds__(128)
void bn_relu_maxpool_kernel(const float* __restrict__ y2, const float* __restrict__ norm2,
                            float* __restrict__ outF) {
    const int bs = blockIdx.x, o = threadIdx.x;
    const float sc = norm2[o], sh = norm2[128 + o];
    float m = -3.4e38f;
    for (int k = 0; k < K_; ++k) {
        const float v = y2[((size_t)bs * K_ + k) * 128 + o];
        m = fmaxf(m, fmaf(v, sc, sh));
    }
    outF[(size_t)bs * 128 + o] = fmaxf(m, 0.f);
}

extern "C" void kernel_launch(void* const* d_in, const int* in_sizes, int n_in,
                              void* d_out, int out_size, void* d_ws, size_t ws_size,
                              hipStream_t stream) {
    const float* xyz  = (const float*)d_in[0];
    const float* pf   = (const float*)d_in[1];
    const float* W0   = (const float*)d_in[2];
    const float* b0   = (const float*)d_in[3];
    const float* g0   = (const float*)d_in[4];
    const float* be0  = (const float*)d_in[5];
    const float* W1   = (const float*)d_in[6];
    const float* b1   = (const float*)d_in[7];
    const float* g1   = (const float*)d_in[8];
    const float* be1  = (const float*)d_in[9];
    const float* W2   = (const float*)d_in[10];
    const float* b2   = (const float*)d_in[11];
    const float* g2   = (const float*)d_in[12];
    const float* be2  = (const float*)d_in[13];

    float* out = (float*)d_out;                   // [B,S,3] then [B,S,128]
    float* ws  = (float*)d_ws;
    float* x0  = ws + OFF_X0;
    float* y0  = ws + OFF_Y0;
    float* y1  = ws + OFF_Y1;
    float* y2  = ws + OFF_Y2;
    float* st0 = ws + OFF_ST0; float* st1 = ws + OFF_ST1; float* st2 = ws + OFF_ST2;
    float* nr0 = ws + OFF_NRM0; float* nr1 = ws + OFF_NRM1; float* nr2 = ws + OFF_NRM2;

    float* sampled = out;                 // [B*S, 3]
    float* feats   = out + (size_t)B_ * S_ * 3;

    zero_kernel<<<1, 512, 0, stream>>>(st0, 512);   // zero all three stat blocks

    fps_kernel<<<B_, 1024, 0, stream>>>(xyz, sampled);

    knn_group_kernel<<<B_ * 64, 256, 0, stream>>>(xyz, pf, sampled, x0);

    const int gblocks = (int)(P_ / 128);
    gemm_bn_kernel<C0IN, C0P, 64, false><<<gblocks, 256, 0, stream>>>(x0, W0, b0, nullptr, y0, st0);
    finalize_stats_kernel<<<1, 64, 0, stream>>>(st0, g0, be0, nr0, 64);

    gemm_bn_kernel<64, 64, 64, true><<<gblocks, 256, 0, stream>>>(y0, W1, b1, nr0, y1, st1);
    finalize_stats_kernel<<<1, 64, 0, stream>>>(st1, g1, be1, nr1, 64);

    gemm_bn_kernel<64, 64, 128, true><<<gblocks, 256, 0, stream>>>(y1, W2, b2, nr1, y2, st2);
    finalize_stats_kernel<<<1, 128, 0, stream>>>(st2, g2, be2, nr2, 128);

    bn_relu_maxpool_kernel<<<B_ * S_, 128, 0, stream>>>(y2, nr2, feats);
}